// GNNSceneEncoder_32220844655119
// MI455X (gfx1250) — compile-verified
//
#include <hip/hip_runtime.h>

typedef __attribute__((ext_vector_type(2))) float v2f;
typedef __attribute__((ext_vector_type(8))) float v8f;

#define N_NODES  50000
#define N_EDGES  800000
#define NODE_DIM 64
#define HIDDEN   128
#define OUT_DIM  256
#define N_LAYERS 3

// ---------------- utility: zero a float buffer ----------------
__global__ void zero_kernel(float* __restrict__ p, int n) {
    int i = blockIdx.x * blockDim.x + threadIdx.x;
    if (i < n) p[i] = 0.0f;
}

// ---------------- h = x @ Wp + bp  (WMMA f32 16x16x4) ----------------
// block = 256 threads = 8 waves; block computes rows [r0, r0+16) x all 128 cols.
// wave w computes the 16x16 tile at columns [16w, 16w+16).
__global__ void proj_kernel(const float* __restrict__ x, const float* __restrict__ Wp,
                            const float* __restrict__ bp, float* __restrict__ h) {
    __shared__ float As[16 * 68];   // 16 x 64 tile of x, padded stride 68 (bank spread)
    const int r0 = blockIdx.x * 16;
    for (int i = threadIdx.x; i < 16 * NODE_DIM; i += 256)
        As[(i >> 6) * 68 + (i & 63)] = x[r0 * NODE_DIM + i];
    __syncthreads();

    const int wave = threadIdx.x >> 5;
    const int lane = threadIdx.x & 31;
    const int n0   = wave * 16;
    const int half = lane >> 4;     // 0: K={0,1}, 1: K={2,3} within a k-step
    const int mn   = lane & 15;     // A row (M) / B col (N)

    v8f c = {};
    for (int kk = 0; kk < NODE_DIM; kk += 4) {
        const int kb = kk + half * 2;
        v2f a, b;
        a.x = As[mn * 68 + kb];
        a.y = As[mn * 68 + kb + 1];
        b.x = Wp[kb * HIDDEN + n0 + mn];
        b.y = Wp[(kb + 1) * HIDDEN + n0 + mn];
        c = __builtin_amdgcn_wmma_f32_16x16x4_f32(false, a, false, b, (short)0, c,
                                                  false, false);
    }
    const int col  = n0 + mn;
    const float bb = bp[col];
#pragma unroll
    for (int v = 0; v < 8; ++v) {
        const int row = r0 + v + half * 8;   // ISA C/D layout: vgpr v -> M = v + 8*half
        h[row * HIDDEN + col] = c[v] + bb;
    }
}

// ---------------- degree / inverse degree ----------------
__global__ void degree_kernel(const int* __restrict__ dst, float* __restrict__ deg) {
    int e = blockIdx.x * blockDim.x + threadIdx.x;
    if (e < N_EDGES) atomicAdd(&deg[dst[e]], 1.0f);
}

__global__ void invdeg_kernel(const float* __restrict__ deg, float* __restrict__ inv) {
    int i = blockIdx.x * blockDim.x + threadIdx.x;
    if (i < N_NODES) {
        float d = deg[i];
        inv[i] = (d > 0.0f) ? (1.0f / d) : 0.0f;
    }
}

// ---------------- edge scatter: msg[dst] += h[src]  ----------------
// one wave per edge; lane handles 4 channels (32 lanes * 4 = 128).
__global__ void scatter_kernel(const int* __restrict__ src, const int* __restrict__ dst,
                               const float* __restrict__ h, float* __restrict__ msg) {
    const int lane = threadIdx.x & 31;
    const int e    = blockIdx.x * (blockDim.x >> 5) + (threadIdx.x >> 5);
    if (e >= N_EDGES) return;
    const int c0 = lane * 4;
    // prefetch a lookahead source row into cache (global_prefetch_b8)
    if (e + 64 < N_EDGES)
        __builtin_prefetch(h + (size_t)src[e + 64] * HIDDEN + c0, 0, 0);
    const int s = src[e], d = dst[e];
    const float4 v = *(const float4*)(h + (size_t)s * HIDDEN + c0);
    float* mp = msg + (size_t)d * HIDDEN + c0;
    atomicAdd(mp + 0, v.x);
    atomicAdd(mp + 1, v.y);
    atomicAdd(mp + 2, v.z);
    atomicAdd(mp + 3, v.w);
}

// ------- h = relu(h + (msg*inv_deg) @ Wg + bg)  (WMMA f32, in place) -------
__global__ void update_kernel(const float* __restrict__ msg, const float* __restrict__ inv_deg,
                              const float* __restrict__ Wg, const float* __restrict__ bg,
                              float* __restrict__ h) {
    __shared__ float Ms[16 * 132];  // 16 x 128 tile, padded stride 132
    const int r0 = blockIdx.x * 16;
    for (int i = threadIdx.x; i < 16 * HIDDEN; i += 256) {
        const int row = i >> 7, col = i & 127;
        Ms[row * 132 + col] = msg[r0 * HIDDEN + i] * inv_deg[r0 + row];
    }
    __syncthreads();

    const int wave = threadIdx.x >> 5;
    const int lane = threadIdx.x & 31;
    const int n0   = wave * 16;
    const int half = lane >> 4;
    const int mn   = lane & 15;

    v8f c = {};
    for (int kk = 0; kk < HIDDEN; kk += 4) {
        const int kb = kk + half * 2;
        v2f a, b;
        a.x = Ms[mn * 132 + kb];
        a.y = Ms[mn * 132 + kb + 1];
        b.x = Wg[kb * HIDDEN + n0 + mn];
        b.y = Wg[(kb + 1) * HIDDEN + n0 + mn];
        c = __builtin_amdgcn_wmma_f32_16x16x4_f32(false, a, false, b, (short)0, c,
                                                  false, false);
    }
    const int col  = n0 + mn;
    const float bb = bg[col];
#pragma unroll
    for (int v = 0; v < 8; ++v) {
        const int row = r0 + v + half * 8;
        const float val = h[row * HIDDEN + col] + c[v] + bb;
        h[row * HIDDEN + col] = val > 0.0f ? val : 0.0f;
    }
}

// ---------------- g = mean(h, axis=0) ----------------
__global__ void mean_kernel(const float* __restrict__ h, float* __restrict__ g) {
    const int col = threadIdx.x & 127;
    const int sub = threadIdx.x >> 7;    // 0..1
    float acc = 0.0f;
    for (int row = blockIdx.x * 2 + sub; row < N_NODES; row += gridDim.x * 2)
        acc += h[(size_t)row * HIDDEN + col];
    atomicAdd(&g[col], acc * (1.0f / (float)N_NODES));
}

// ---------------- latent = relu(g@W1+b1)@W2+b2 ----------------
__global__ void mlp_kernel(const float* __restrict__ g, const float* __restrict__ W1,
                           const float* __restrict__ b1, const float* __restrict__ W2,
                           const float* __restrict__ b2, float* __restrict__ out) {
    __shared__ float gg[HIDDEN];
    __shared__ float t[HIDDEN];
    if (threadIdx.x < HIDDEN) gg[threadIdx.x] = g[threadIdx.x];
    __syncthreads();
    if (threadIdx.x < HIDDEN) {
        float acc = b1[threadIdx.x];
        for (int k = 0; k < HIDDEN; ++k) acc += gg[k] * W1[k * HIDDEN + threadIdx.x];
        t[threadIdx.x] = acc > 0.0f ? acc : 0.0f;
    }
    __syncthreads();
    float acc = b2[threadIdx.x];
    for (int j = 0; j < HIDDEN; ++j) acc += t[j] * W2[j * OUT_DIM + threadIdx.x];
    out[threadIdx.x] = acc;
}

extern "C" void kernel_launch(void* const* d_in, const int* in_sizes, int n_in,
                              void* d_out, int out_size, void* d_ws, size_t ws_size,
                              hipStream_t stream) {
    const float* x  = (const float*)d_in[0];
    const int*   src = (const int*)d_in[1];
    const int*   dst = (const int*)d_in[2];
    const float* Wp = (const float*)d_in[3];
    const float* bp = (const float*)d_in[4];
    const float* Wg = (const float*)d_in[5];
    const float* bg = (const float*)d_in[6];
    const float* W1 = (const float*)d_in[7];
    const float* b1 = (const float*)d_in[8];
    const float* W2 = (const float*)d_in[9];
    const float* b2 = (const float*)d_in[10];
    float* out = (float*)d_out;

    // workspace layout (floats): h[6.4M] | msg[6.4M] | deg[50k] | inv[50k] | g[128]
    float* h   = (float*)d_ws;
    float* msg = h + (size_t)N_NODES * HIDDEN;
    float* deg = msg + (size_t)N_NODES * HIDDEN;
    float* inv = deg + N_NODES;
    float* g   = inv + N_NODES;

    const int HN = N_NODES * HIDDEN;

    // degree + inverse degree
    zero_kernel<<<(N_NODES + 255) / 256, 256, 0, stream>>>(deg, N_NODES);
    degree_kernel<<<(N_EDGES + 255) / 256, 256, 0, stream>>>(dst, deg);
    invdeg_kernel<<<(N_NODES + 255) / 256, 256, 0, stream>>>(deg, inv);

    // node projection (3125 blocks x 16 rows = 50000 exactly)
    proj_kernel<<<N_NODES / 16, 256, 0, stream>>>(x, Wp, bp, h);

    for (int l = 0; l < N_LAYERS; ++l) {
        zero_kernel<<<(HN + 255) / 256, 256, 0, stream>>>(msg, HN);
        scatter_kernel<<<N_EDGES / 8, 256, 0, stream>>>(src, dst, h, msg);
        update_kernel<<<N_NODES / 16, 256, 0, stream>>>(
            msg, inv, Wg + (size_t)l * HIDDEN * HIDDEN, bg + (size_t)l * HIDDEN, h);
    }

    zero_kernel<<<1, 128, 0, stream>>>(g, HIDDEN);
    mean_kernel<<<256, 256, 0, stream>>>(h, g);
    mlp_kernel<<<1, 256, 0, stream>>>(g, W1, b1, W2, b2, out);
}